// Memory_55516747268372
// MI455X (gfx1250) — compile-verified
//
#include <hip/hip_runtime.h>
#include <math.h>

typedef __attribute__((ext_vector_type(2))) float v2f;
typedef __attribute__((ext_vector_type(8))) float v8f;

#define T_      4
#define NS_     25
#define NQ_     75
#define NN_     100         // rows per t after concat
#define ROWS_   400         // T_ * NN_
#define D_      128
#define M_      512
#define EPS_    1e-12f
#define MARGIN_ 0.1f

__device__ __forceinline__ float waveSum(float v) {
#pragma unroll
    for (int off = 16; off > 0; off >>= 1)
        v += __shfl_xor(v, off, 32);
    return v;
}

// ---------------------------------------------------------------------------
// K0: L2-normalize the concatenated embeddings.
// waves [0,400)  -> norm_emb  (written straight into d_out, it is output #1)
// waves [400,800)-> norm_emb_glo (workspace)
// one wave per 128-float row; float4 per lane.
// ---------------------------------------------------------------------------
__global__ void k_normalize_inputs(const float* __restrict__ emb_s,
                                   const float* __restrict__ emb_q,
                                   const float* __restrict__ embg_s,
                                   const float* __restrict__ embg_q,
                                   float* __restrict__ norm_emb,
                                   float* __restrict__ norm_glo) {
    int wave = (blockIdx.x * blockDim.x + threadIdx.x) >> 5;
    int lane = threadIdx.x & 31;
    int glo  = wave >= ROWS_;
    int rr   = glo ? wave - ROWS_ : wave;
    int t    = rr / NN_;
    int n    = rr % NN_;
    const float* src;
    if (!glo)
        src = (n < NS_) ? emb_s  + (t * NS_ + n) * D_ : emb_q  + (t * NQ_ + (n - NS_)) * D_;
    else
        src = (n < NS_) ? embg_s + (t * NS_ + n) * D_ : embg_q + (t * NQ_ + (n - NS_)) * D_;
    float4 v = *(const float4*)(src + lane * 4);
    float ss = waveSum(v.x * v.x + v.y * v.y + v.z * v.z + v.w * v.w);
    float inv = 1.0f / fmaxf(sqrtf(ss), EPS_);
    v.x *= inv; v.y *= inv; v.z *= inv; v.w *= inv;
    float* dst = (glo ? norm_glo : norm_emb) + rr * D_;
    *(float4*)(dst + lane * 4) = v;
}

// ---------------------------------------------------------------------------
// K1: sim_kv = norm_emb @ keys^T   [400x512]
//     sim_vk = norm_glo @ values^T [400x512]
// One wave per 16x16 output tile, fp32 WMMA, K stepped by 4 (32 wmma/tile).
// Tiles: 25 row-tiles * 32 col-tiles * 2 sims = 1600 waves = 200 blocks.
// A frag: lane<16 -> K={k,k+1}, lane>=16 -> K={k+2,k+3}, M = lane&15 (+tile).
// B frag mirrors per column. B[k][col] = mat[col*128 + k] (row-major over D).
// ---------------------------------------------------------------------------
__global__ void k_sim_wmma(const float* __restrict__ norm_emb,
                           const float* __restrict__ norm_glo,
                           const float* __restrict__ keys,
                           const float* __restrict__ values,
                           float* __restrict__ sim_kv,
                           float* __restrict__ sim_vk) {
    int wave = (blockIdx.x * blockDim.x + threadIdx.x) >> 5;
    int lane = threadIdx.x & 31;
    int s    = wave >= 800;
    int tile = s ? wave - 800 : wave;
    int rt = tile >> 5;          // 0..24
    int ct = tile & 31;          // 0..31
    const float* A = s ? norm_glo : norm_emb;
    const float* B = s ? values  : keys;
    float*       S = s ? sim_vk  : sim_kv;

    int row = rt * 16 + (lane & 15);
    int col = ct * 16 + (lane & 15);
    int klo = (lane >> 4) << 1;          // 0 or 2
    const float* arow = A + row * D_ + klo;
    const float* bcol = B + col * D_ + klo;

    v8f acc = {0.f, 0.f, 0.f, 0.f, 0.f, 0.f, 0.f, 0.f};
#pragma unroll
    for (int k = 0; k < D_; k += 4) {
        v2f a = *(const v2f*)(arow + k);
        v2f b = *(const v2f*)(bcol + k);
        acc = __builtin_amdgcn_wmma_f32_16x16x4_f32(false, a, false, b,
                                                    (short)0, acc, false, false);
    }
    int hi = (lane >> 4) * 8;
    float* out = S + (rt * 16 + hi) * M_ + col;
#pragma unroll
    for (int r = 0; r < 8; ++r)
        out[r * M_] = acc[r];
}

// ---------------------------------------------------------------------------
// K2: per-row masked argmin (hardest positive) / argmax (hardest negative)
// for both sims, any-flags, and per-row sum((sim_vk - sim_kv)^2).
// One wave per row (400 waves); lane scans 16 strided columns; shfl reduce
// with index tie-break = first occurrence (matches jnp.argmin/argmax).
// ---------------------------------------------------------------------------
__global__ void k_row_stats(const float* __restrict__ sim_kv,
                            const float* __restrict__ sim_vk,
                            const float* __restrict__ thresh,
                            int* __restrict__ minpos_kv, int* __restrict__ maxneg_kv,
                            int* __restrict__ minpos_vk, int* __restrict__ maxneg_vk,
                            float* __restrict__ poss_kv, float* __restrict__ negs_kv,
                            float* __restrict__ poss_vk, float* __restrict__ negs_vk,
                            float* __restrict__ rowSumSq) {
    int wave = (blockIdx.x * blockDim.x + threadIdx.x) >> 5;
    int lane = threadIdx.x & 31;
    int n = wave;                          // 0..399
    float tp0 = thresh[0], tn0 = thresh[1], tp1 = thresh[2], tn1 = thresh[3];
    const float* rk = sim_kv + n * M_;
    const float* rv = sim_vk + n * M_;

    float minv0 =  INFINITY, minv1 =  INFINITY;
    float maxv0 = -INFINITY, maxv1 = -INFINITY;
    int   mini0 = 0, mini1 = 0, maxi0 = 0, maxi1 = 0;
    float ssq = 0.f;
#pragma unroll
    for (int i = 0; i < 16; ++i) {
        int c = lane + i * 32;             // increasing per lane -> first-occurrence
        float a = rk[c];
        float b = rv[c];
        float d = b - a;
        ssq += d * d;
        if (a > tp0 && a < minv0) { minv0 = a; mini0 = c; }
        if (a < tn0 && a > maxv0) { maxv0 = a; maxi0 = c; }
        if (b > tp1 && b < minv1) { minv1 = b; mini1 = c; }
        if (b < tn1 && b > maxv1) { maxv1 = b; maxi1 = c; }
    }
#pragma unroll
    for (int off = 16; off > 0; off >>= 1) {
        float ov; int oi;
        ov = __shfl_xor(minv0, off, 32); oi = __shfl_xor(mini0, off, 32);
        if (ov < minv0 || (ov == minv0 && oi < mini0)) { minv0 = ov; mini0 = oi; }
        ov = __shfl_xor(maxv0, off, 32); oi = __shfl_xor(maxi0, off, 32);
        if (ov > maxv0 || (ov == maxv0 && oi < maxi0)) { maxv0 = ov; maxi0 = oi; }
        ov = __shfl_xor(minv1, off, 32); oi = __shfl_xor(mini1, off, 32);
        if (ov < minv1 || (ov == minv1 && oi < mini1)) { minv1 = ov; mini1 = oi; }
        ov = __shfl_xor(maxv1, off, 32); oi = __shfl_xor(maxi1, off, 32);
        if (ov > maxv1 || (ov == maxv1 && oi < maxi1)) { maxv1 = ov; maxi1 = oi; }
    }
    ssq = waveSum(ssq);
    if (lane == 0) {
        minpos_kv[n] = mini0; maxneg_kv[n] = maxi0;
        minpos_vk[n] = mini1; maxneg_vk[n] = maxi1;
        poss_kv[n] = (minv0 !=  INFINITY) ?  1.f : 0.f;
        negs_kv[n] = (maxv0 != -INFINITY) ? -1.f : 0.f;
        poss_vk[n] = (minv1 !=  INFINITY) ?  1.f : 0.f;
        negs_vk[n] = (maxv1 != -INFINITY) ? -1.f : 0.f;
        rowSumSq[n] = ssq;
    }
}

// ---------------------------------------------------------------------------
// K3: gather hardest pos/neg vectors and form per-row loss contributions.
// rowLossV[n] = pos_s*dot(values[minpos], glo[n]) + neg_s*dot(values[maxneg], glo[n])
// rowLossK[n] = analogous with keys vs norm_emb.
// One wave per row, 4 dims per lane.
// ---------------------------------------------------------------------------
__global__ void k_row_loss(const float* __restrict__ norm_emb,
                           const float* __restrict__ norm_glo,
                           const float* __restrict__ keys,
                           const float* __restrict__ values,
                           const int* __restrict__ minpos_kv, const int* __restrict__ maxneg_kv,
                           const int* __restrict__ minpos_vk, const int* __restrict__ maxneg_vk,
                           const float* __restrict__ poss_kv, const float* __restrict__ negs_kv,
                           const float* __restrict__ poss_vk, const float* __restrict__ negs_vk,
                           float* __restrict__ rowLossV, float* __restrict__ rowLossK) {
    int wave = (blockIdx.x * blockDim.x + threadIdx.x) >> 5;
    int lane = threadIdx.x & 31;
    int n = wave;
    float4 g = *(const float4*)(norm_glo + n * D_ + lane * 4);
    float4 e = *(const float4*)(norm_emb + n * D_ + lane * 4);
    float4 a;
    a = *(const float4*)(values + minpos_kv[n] * D_ + lane * 4);
    float d1 = a.x * g.x + a.y * g.y + a.z * g.z + a.w * g.w;
    a = *(const float4*)(values + maxneg_kv[n] * D_ + lane * 4);
    float d2 = a.x * g.x + a.y * g.y + a.z * g.z + a.w * g.w;
    a = *(const float4*)(keys + minpos_vk[n] * D_ + lane * 4);
    float d3 = a.x * e.x + a.y * e.y + a.z * e.z + a.w * e.w;
    a = *(const float4*)(keys + maxneg_vk[n] * D_ + lane * 4);
    float d4 = a.x * e.x + a.y * e.y + a.z * e.z + a.w * e.w;
    d1 = waveSum(d1); d2 = waveSum(d2); d3 = waveSum(d3); d4 = waveSum(d4);
    if (lane == 0) {
        rowLossV[n] = poss_kv[n] * d1 + negs_kv[n] * d2;
        rowLossK[n] = poss_vk[n] * d3 + negs_vk[n] * d4;
    }
}

// ---------------------------------------------------------------------------
// K4: pre = norm_glo + pos_score @ values   [400x128], fp32 WMMA.
// pos_score[n][m] = sim_kv[n][m] > thresh0 ? sim_kv[n][m] : 0 (applied on load).
// 25 row-tiles * 8 col-tiles = 200 waves = 25 blocks; K=512 -> 128 wmma/tile.
// ---------------------------------------------------------------------------
__global__ void k_glo_out_wmma(const float* __restrict__ sim_kv,
                               const float* __restrict__ values,
                               const float* __restrict__ norm_glo,
                               const float* __restrict__ thresh,
                               float* __restrict__ pre) {
    int wave = (blockIdx.x * blockDim.x + threadIdx.x) >> 5;
    int lane = threadIdx.x & 31;
    float tp = thresh[0];
    int rt = wave >> 3;          // 0..24
    int ct = wave & 7;           // 0..7
    int row = rt * 16 + (lane & 15);
    int col = ct * 16 + (lane & 15);
    int klo = (lane >> 4) << 1;
    const float* arow = sim_kv + row * M_ + klo;
    const float* bcol = values + klo * D_ + col;

    v8f acc = {0.f, 0.f, 0.f, 0.f, 0.f, 0.f, 0.f, 0.f};
#pragma unroll 8
    for (int k = 0; k < M_; k += 4) {
        v2f a = *(const v2f*)(arow + k);
        a.x = (a.x > tp) ? a.x : 0.f;
        a.y = (a.y > tp) ? a.y : 0.f;
        v2f b;
        b.x = bcol[k * D_];            // values[(k+klo  )*128 + col]
        b.y = bcol[k * D_ + D_];       // values[(k+klo+1)*128 + col]
        acc = __builtin_amdgcn_wmma_f32_16x16x4_f32(false, a, false, b,
                                                    (short)0, acc, false, false);
    }
    int hi = (lane >> 4) * 8;
#pragma unroll
    for (int r = 0; r < 8; ++r) {
        int rrow = rt * 16 + hi + r;
        pre[rrow * D_ + col] = acc[r] + norm_glo[rrow * D_ + col];
    }
}

// ---------------------------------------------------------------------------
// K5: row-normalize pre -> emb_glo_out (output #2), and one extra wave
// deterministically folds the per-row loss arrays into the 3 scalar outputs.
// ---------------------------------------------------------------------------
__global__ void k_finalize(const float* __restrict__ pre,
                           const float* __restrict__ rowSumSq,
                           const float* __restrict__ rowLossV,
                           const float* __restrict__ rowLossK,
                           float* __restrict__ out_glo,
                           float* __restrict__ out_losses) {
    int wave = (blockIdx.x * blockDim.x + threadIdx.x) >> 5;
    int lane = threadIdx.x & 31;
    if (wave < ROWS_) {
        float4 v = *(const float4*)(pre + wave * D_ + lane * 4);
        float ss = waveSum(v.x * v.x + v.y * v.y + v.z * v.z + v.w * v.w);
        float inv = 1.0f / fmaxf(sqrtf(ss), EPS_);
        v.x *= inv; v.y *= inv; v.z *= inv; v.w *= inv;
        *(float4*)(out_glo + wave * D_ + lane * 4) = v;
    } else if (wave == ROWS_) {
        float sS = 0.f, sV = 0.f, sK = 0.f;
        for (int i = lane; i < ROWS_; i += 32) {
            sS += rowSumSq[i];
            sV += rowLossV[i];
            sK += rowLossK[i];
        }
        sS = waveSum(sS); sV = waveSum(sV); sK = waveSum(sK);
        if (lane == 0) {
            out_losses[0] = fmaxf(-(sK * (1.0f / ROWS_)) + MARGIN_, 0.f);   // loss_k
            out_losses[1] = fmaxf(-(sV * (1.0f / ROWS_)) + MARGIN_, 0.f);   // loss_v
            out_losses[2] = sS * (1.0f / (ROWS_ * (float)M_));              // loss_s
        }
    }
}

extern "C" void kernel_launch(void* const* d_in, const int* in_sizes, int n_in,
                              void* d_out, int out_size, void* d_ws, size_t ws_size,
                              hipStream_t stream) {
    const float* emb_s  = (const float*)d_in[0];
    const float* emb_q  = (const float*)d_in[1];
    const float* embg_s = (const float*)d_in[2];
    const float* embg_q = (const float*)d_in[3];
    const float* keys   = (const float*)d_in[4];
    const float* values = (const float*)d_in[5];
    const float* thresh = (const float*)d_in[6];
    float* out = (float*)d_out;                  // [0,51200) norm_emb | [51200,102400) emb_glo_out | 3 losses
    float* w   = (float*)d_ws;

    float* glo      = w;                 // 51200
    float* sim_kv   = w + 51200;         // 204800
    float* sim_vk   = w + 256000;        // 204800
    float* pre      = w + 460800;        // 51200
    int*   iw       = (int*)(w + 512000);
    int* minpos_kv  = iw;        int* maxneg_kv = iw + 400;
    int* minpos_vk  = iw + 800;  int* maxneg_vk = iw + 1200;
    float* poss_kv  = w + 513600; float* negs_kv = w + 514000;
    float* poss_vk  = w + 514400; float* negs_vk = w + 514800;
    float* rowSumSq = w + 515200;
    float* rowLossV = w + 515600;
    float* rowLossK = w + 516000;

    k_normalize_inputs<<<100, 256, 0, stream>>>(emb_s, emb_q, embg_s, embg_q, out, glo);
    k_sim_wmma<<<200, 256, 0, stream>>>(out, glo, keys, values, sim_kv, sim_vk);
    k_row_stats<<<50, 256, 0, stream>>>(sim_kv, sim_vk, thresh,
                                        minpos_kv, maxneg_kv, minpos_vk, maxneg_vk,
                                        poss_kv, negs_kv, poss_vk, negs_vk, rowSumSq);
    k_row_loss<<<50, 256, 0, stream>>>(out, glo, keys, values,
                                       minpos_kv, maxneg_kv, minpos_vk, maxneg_vk,
                                       poss_kv, negs_kv, poss_vk, negs_vk,
                                       rowLossV, rowLossK);
    k_glo_out_wmma<<<25, 256, 0, stream>>>(sim_kv, values, glo, thresh, pre);
    k_finalize<<<51, 256, 0, stream>>>(pre, rowSumSq, rowLossV, rowLossK,
                                       out + 51200, out + 102400);
}